// AdaptivePatchEmbedding_76811195121893
// MI455X (gfx1250) — compile-verified
//
#include <hip/hip_runtime.h>
#include <math.h>

// ---- problem constants ----
#define NB      128
#define NC      32
#define NS      336
#define NREG    28672          // (NB*NC) * (NS/48)
#define RP      7              // regions per series
#define NN      4096           // NB*NC
#define TGT     6              // patches per region
#define DM      512            // d_model
#define MAXP    48
#define XPATCH_ELEMS (NN * (RP*TGT) * DM)   // 88,080,384

typedef float v2f __attribute__((ext_vector_type(2)));
typedef float v8f __attribute__((ext_vector_type(8)));

// ---- CDNA5 async-to-LDS helpers ----
__device__ __forceinline__ void async_ld_b32(unsigned ldsAddr, const float* gaddr) {
    // GLOBAL_LOAD_ASYNC_TO_LDS_B32, GV mode: vdst = LDS byte address,
    // vaddr = 64-bit global address. Tracked by ASYNCcnt.
    asm volatile("global_load_async_to_lds_b32 %0, %1, off"
                 :: "v"(ldsAddr), "v"(gaddr) : "memory");
}
__device__ __forceinline__ void wait_asynccnt0() {
#if __has_builtin(__builtin_amdgcn_s_wait_asynccnt)
    __builtin_amdgcn_s_wait_asynccnt(0);
#else
    asm volatile("s_wait_asynccnt 0x0" ::: "memory");
#endif
}

// ============================================================
// Kernel 1: router MLP + gumbel argmax (one thread per region)
// ============================================================
__global__ __launch_bounds__(256) void router_kernel(
    const float* __restrict__ x, const float* __restrict__ u_noise,
    const float* __restrict__ W1, const float* __restrict__ b1,
    const float* __restrict__ W2, const float* __restrict__ b2,
    int* __restrict__ experts, float* __restrict__ out)
{
    __shared__ float sW1[48 * 64];
    __shared__ float sB1[64];
    __shared__ float sW2[64 * 4];
    __shared__ float sB2[4];

    int tid = threadIdx.x;
    for (int i = tid; i < 48 * 64; i += 256) sW1[i] = W1[i];
    for (int i = tid; i < 64 * 4;  i += 256) sW2[i] = W2[i];
    if (tid < 64) sB1[tid] = b1[tid];
    if (tid < 4)  sB2[tid] = b2[tid];
    __syncthreads();

    int g = blockIdx.x * 256 + tid;          // region index = n*7 + r
    float xr[48];
    #pragma unroll
    for (int k = 0; k < 48; ++k) xr[k] = x[g * 48 + k];

    float lg0 = sB2[0], lg1 = sB2[1], lg2 = sB2[2], lg3 = sB2[3];
    #pragma unroll 4
    for (int j = 0; j < 64; ++j) {
        float h = sB1[j];
        #pragma unroll
        for (int k = 0; k < 48; ++k) h = fmaf(xr[k], sW1[k * 64 + j], h);
        h = fmaxf(h, 0.0f);
        lg0 = fmaf(h, sW2[j * 4 + 0], lg0);
        lg1 = fmaf(h, sW2[j * 4 + 1], lg1);
        lg2 = fmaf(h, sW2[j * 4 + 2], lg2);
        lg3 = fmaf(h, sW2[j * 4 + 3], lg3);
    }

    float lg[4] = {lg0, lg1, lg2, lg3};
    #pragma unroll
    for (int m = 0; m < 4; ++m) {
        float u = u_noise[g * 4 + m];
        float gum = -__logf(-__logf(u + 1e-10f) + 1e-10f);
        lg[m] += gum;
    }
    int e = 0; float best = lg[0];
    #pragma unroll
    for (int m = 1; m < 4; ++m) { if (lg[m] > best) { best = lg[m]; e = m; } }

    experts[g] = e;
    int n = g / RP, r = g - n * RP;
    // cls_pred: argmax[N,R].T.reshape(-1) -> index r*N + n, after x_patch + C
    out[XPATCH_ELEMS + 1 + r * NN + n] = (float)e;
    if (g == 0) out[XPATCH_ELEMS] = 32.0f;   // jnp.int32(C)
}

// ============================================================
// Kernel 2: positional embedding table (42 x 512), fp64 math
// ============================================================
__global__ __launch_bounds__(256) void posemb_kernel(float* __restrict__ pe)
{
    int idx = blockIdx.x * 256 + threadIdx.x;
    if (idx >= (RP * TGT) * DM) return;
    int p = idx >> 9;          // position 0..41
    int d = idx & 511;         // channel
    int i2 = d & ~1;           // 2*i
    double div = exp(-(double)i2 * (log(10000.0) / (double)DM));
    double v = (double)p * div;
    pe[idx] = (float)((d & 1) ? cos(v) : sin(v));
}

// ============================================================
// Kernel 3: WMMA patch embedding with double-buffered
//           async-to-LDS tile streaming.
//   Workgroup: 256 thr (8 waves) owns 16 regions; each wave owns 2.
//   Per region: A = 16x48 fp32 (patch rows at M = 0..2 and 8..10,
//   zero-padded, hoisted to VGPRs), B = expert W_emb zero-padded to
//   48 x 16 column tile in LDS, 12x v_wmma_f32_16x16x4_f32 with a
//   1-deep software-pipelined B-fragment load. W/PE column tiles for
//   ct+1 stream via GLOBAL_LOAD_ASYNC_TO_LDS_B32 into the alternate
//   LDS buffer while tile ct is consumed. Output stores use hoisted
//   per-region base pointers + immediate offsets.
// ============================================================
#define TILE_F   3744          // floats per buffer: 4*48*16 (W) + 42*16 (PE)
#define W_VALID  1536          // valid (non-pad) W dwords per tile: (8+16+24+48)*16
#define TILE_LD  2208          // async dwords per tile: W_VALID + 42*16

__global__ __launch_bounds__(256) void embed_kernel(
    const float* __restrict__ x, const int* __restrict__ experts,
    const float* __restrict__ pe,
    const float* __restrict__ We0, const float* __restrict__ We1,
    const float* __restrict__ We2, const float* __restrict__ We3,
    float* __restrict__ out)
{
    __shared__ float sX[16 * 48];        // 16 region rows
    __shared__ int   sE[16];
    __shared__ float sBuf[2][TILE_F];    // double-buffered W + PE tiles

    int tid  = threadIdx.x;
    int lane = tid & 31;
    int wave = tid >> 5;
    int rb   = blockIdx.x * 16;          // first region of this WG

    // zero both buffers once: pad rows (k >= plen) stay 0 forever,
    // async loads only ever fill the valid rows + PE.
    for (int i = tid; i < 2 * TILE_F; i += 256) (&sBuf[0][0])[i] = 0.0f;
    for (int i = tid; i < 16 * 48; i += 256) sX[i] = x[rb * 48 + i];
    if (tid < 16) sE[tid] = experts[rb + tid];

    // ---- per-thread async-load assignments (max 9 each) ----
    unsigned ldsOff[9];                  // LDS byte addr in buffer 0
    const float* gb[9];                  // global addr for ct = 0
    #pragma unroll
    for (int a = 0; a < 9; ++a) {
        unsigned i = (unsigned)tid + a * 256u;
        unsigned lo = 0; const float* g = nullptr;
        if (i < W_VALID) {
            unsigned r = i >> 4, d = i & 15, e, k;
            if (r < 8)       { e = 0; k = r; }
            else if (r < 24) { e = 1; k = r - 8; }
            else if (r < 48) { e = 2; k = r - 24; }
            else             { e = 3; k = r - 48; }
            const float* W = (e == 0) ? We0 : (e == 1) ? We1
                           : (e == 2) ? We2 : We3;
            g  = W + k * DM + d;
            lo = (unsigned)(uintptr_t)&sBuf[0][e * 768 + k * 16 + d];
        } else if (i < TILE_LD) {
            unsigned j = i - W_VALID;
            g  = pe + (j >> 4) * DM + (j & 15);
            lo = (unsigned)(uintptr_t)&sBuf[0][3072 + j];
        }
        ldsOff[a] = lo; gb[a] = g;
    }

    __syncthreads();                     // zeros + sX/sE visible

    // kick off tile 0 into buffer 0
    #pragma unroll
    for (int a = 0; a < 9; ++a)
        if ((unsigned)tid + a * 256u < TILE_LD) async_ld_b32(ldsOff[a], gb[a]);

    // ---- per-lane A-fragment geometry ----
    int row = lane & 15;                 // M row (and B column)
    int kb  = (lane >> 4) * 2;           // K sub-offset within 4-wide step
    // patch slot t: rows 0..2 -> t=0..2, rows 8..10 -> t=3..5
    int t = (row < 3) ? row : ((row >= 8 && row < 11) ? (row - 5) : -1);
    int pbase = (lane < 16) ? 0 : 3;     // C-fragment patch base per half-wave

    // ---- hoist A fragments, output bases, PE offsets for 2 regions ----
    v2f aval[2][12];
    int eReg[2];
    float* oBase[2];                     // &out[(n*42 + r*6 + pbase)*512 + row]
    int    peOff[2];                     // LDS float idx of PE row (pbase) col row
    #pragma unroll
    for (int rr = 0; rr < 2; ++rr) {
        int lr = wave * 2 + rr;
        int e  = sE[lr];
        eReg[rr] = e;
        int plen = (e == 0) ? 8 : (e == 1) ? 16 : (e == 2) ? 24 : 48;
        int off = 0;
        if (t >= 0) {
            if      (e == 0) off = t * 8;                 // idx = t
            else if (e == 1) off = (t >= 4) ? 16 : 0;     // idx 0,0,0,0,1,1
            else if (e == 2) off = (t == 5) ? 24 : 0;     // idx 0,0,0,0,0,1
            else             off = 0;                     // idx all 0
        }
        #pragma unroll
        for (int kk = 0; kk < 12; ++kk) {
            int k0 = kk * 4 + kb;
            float a0 = 0.0f, a1 = 0.0f;
            if (t >= 0) {
                if (k0     < plen) a0 = sX[lr * 48 + off + k0];
                if (k0 + 1 < plen) a1 = sX[lr * 48 + off + k0 + 1];
            }
            aval[rr][kk].x = a0;
            aval[rr][kk].y = a1;
        }
        int g = rb + lr;
        int n = g / RP, r = g - n * RP;
        oBase[rr] = out + ((size_t)(n * 42 + r * TGT + pbase)) * DM + row;
        peOff[rr] = 3072 + (r * TGT + pbase) * 16 + row;
    }

    // ---- loop over 32 column tiles of D_MODEL ----
    for (int ct = 0; ct < 32; ++ct) {
        int buf = ct & 1;
        wait_asynccnt0();                // my tile-ct loads have landed
        __syncthreads();                 // all waves' loads landed AND all
                                         // waves done reading buf^1
        if (ct + 1 < 32) {               // stream tile ct+1 into buf^1
            unsigned bufd = (unsigned)(buf ^ 1) * (TILE_F * 4u);
            int nxt = (ct + 1) * 16;
            #pragma unroll
            for (int a = 0; a < 9; ++a)
                if ((unsigned)tid + a * 256u < TILE_LD)
                    async_ld_b32(ldsOff[a] + bufd, gb[a] + nxt);
        }

        const float* base = &sBuf[buf][0];
        #pragma unroll
        for (int rr = 0; rr < 2; ++rr) {
            const float* bBase = base + eReg[rr] * 768;
            v8f acc = {0.f, 0.f, 0.f, 0.f, 0.f, 0.f, 0.f, 0.f};
            // 1-deep software pipeline: keep one ds_load in flight past
            // each WMMA (avoids s_wait_dscnt 0x0 on every step).
            v2f bcur;
            bcur.x = bBase[kb * 16 + row];
            bcur.y = bBase[(kb + 1) * 16 + row];
            #pragma unroll
            for (int kk = 0; kk < 12; ++kk) {
                v2f bnxt = bcur;
                if (kk < 11) {
                    bnxt.x = bBase[((kk + 1) * 4 + kb)     * 16 + row];
                    bnxt.y = bBase[((kk + 1) * 4 + kb + 1) * 16 + row];
                }
                acc = __builtin_amdgcn_wmma_f32_16x16x4_f32(
                        false, aval[rr][kk], false, bcur,
                        (short)0, acc, false, false);
                bcur = bnxt;
            }
            // C/D layout: VGPR i -> M = i + 8*(lane>>4); rows 0..2 / 8..10
            // hold patches 0..2 / 3..5 -> all 32 lanes store.
            float* dst = oBase[rr] + ct * 16;
            const float* pebase = base + peOff[rr];
            #pragma unroll
            for (int i = 0; i < 3; ++i)
                dst[i * DM] = acc[i] + pebase[i * 16];
        }
    }
}

// ============================================================
extern "C" void kernel_launch(void* const* d_in, const int* in_sizes, int n_in,
                              void* d_out, int out_size, void* d_ws, size_t ws_size,
                              hipStream_t stream) {
    const float* x   = (const float*)d_in[0];
    const float* u   = (const float*)d_in[1];
    const float* W1  = (const float*)d_in[2];
    const float* b1  = (const float*)d_in[3];
    const float* W2  = (const float*)d_in[4];
    const float* b2  = (const float*)d_in[5];
    const float* We0 = (const float*)d_in[6];
    const float* We1 = (const float*)d_in[7];
    const float* We2 = (const float*)d_in[8];
    const float* We3 = (const float*)d_in[9];
    float* out = (float*)d_out;

    // workspace: [0,112KB) expert ids (28672 ints), [128KB, ...) pos-emb table
    int*   experts = (int*)d_ws;
    float* pe      = (float*)((char*)d_ws + 128 * 1024);

    router_kernel<<<NREG / 256, 256, 0, stream>>>(x, u, W1, b1, W2, b2,
                                                  experts, out);
    posemb_kernel<<<((RP * TGT) * DM + 255) / 256, 256, 0, stream>>>(pe);
    embed_kernel<<<NREG / 16, 256, 0, stream>>>(x, experts, pe,
                                                We0, We1, We2, We3, out);
}